// LinearAutoDecoder_72713796321294
// MI455X (gfx1250) — compile-verified
//
#include <hip/hip_runtime.h>

#define POS_SIZE 63
#define LATENT_SIZE 256
#define NUM_CLUSTERS 64
#define KDIM 319   // POS_SIZE + LATENT_SIZE
#define CW 8       // waves per compute block (256 threads, wave32)

typedef __attribute__((ext_vector_type(2))) float v2f;
typedef __attribute__((ext_vector_type(8))) float v8f;

// ---------------- pass 0: zero cluster counts ----------------
__global__ void init_counts(int* __restrict__ cnt) {
    int t = threadIdx.x;
    if (t < NUM_CLUSTERS) cnt[t] = 0;
}

// ---------------- pass 1: histogram of cluster ids ----------------
__global__ void hist_kernel(const int* __restrict__ cid, int* __restrict__ cnt, int n) {
    __shared__ int h[NUM_CLUSTERS];
    int t = threadIdx.x;
    if (t < NUM_CLUSTERS) h[t] = 0;
    __syncthreads();
    int i = blockIdx.x * blockDim.x + t;
    if (i < n) atomicAdd(&h[cid[i]], 1);
    __syncthreads();
    if (t < NUM_CLUSTERS && h[t] > 0) atomicAdd(&cnt[t], h[t]);
}

// ---------------- pass 2: tiny serial scan (64 entries) ----------------
__global__ void scan_kernel(const int* __restrict__ cnt, int* __restrict__ rowStart,
                            int* __restrict__ chunkStart, int* __restrict__ cursor) {
    if (threadIdx.x == 0 && blockIdx.x == 0) {
        int rs = 0, cs = 0;
        for (int c = 0; c < NUM_CLUSTERS; ++c) {
            rowStart[c] = rs; chunkStart[c] = cs; cursor[c] = rs;
            rs += cnt[c];
            cs += (cnt[c] + 15) >> 4;
        }
        rowStart[NUM_CLUSTERS] = rs;
        chunkStart[NUM_CLUSTERS] = cs;
    }
}

// ---------------- pass 3: scatter row indices into cluster bins ----------------
__global__ void scatter_kernel(const int* __restrict__ cid, int* __restrict__ cursor,
                               int* __restrict__ idx, int n) {
    __shared__ int lh[NUM_CLUSTERS];
    __shared__ int lbase[NUM_CLUSTERS];
    int t = threadIdx.x;
    if (t < NUM_CLUSTERS) lh[t] = 0;
    __syncthreads();
    int i = blockIdx.x * blockDim.x + t;
    int c = 0, my = 0;
    if (i < n) { c = cid[i]; my = atomicAdd(&lh[c], 1); }
    __syncthreads();
    if (t < NUM_CLUSTERS) {
        int k = lh[t];
        lbase[t] = (k > 0) ? atomicAdd(&cursor[t], k) : 0;
    }
    __syncthreads();
    if (i < n) idx[lbase[c] + my] = i;
}

// ---------------- pass 4: per-cluster WMMA: D = W(3x319 pad 16) * X^T(319x16) ----------------
__global__ void __launch_bounds__(256)
compute_kernel(const float* __restrict__ X, const float* __restrict__ Wp,
               const float* __restrict__ Wf, const int* __restrict__ rowStart,
               const int* __restrict__ chunkStart, const int* __restrict__ idx,
               float* __restrict__ out) {
    // Per-wave A-fragment table: 160 K-pairs x 4 cols (cols 0..2 = weight rows, col 3 = zeros)
    __shared__ v2f W4[CW][640];

    const int lane = threadIdx.x & 31;
    const int w    = threadIdx.x >> 5;
    const int seg  = blockIdx.x * CW + w;
    const int total = chunkStart[NUM_CLUSTERS];
    if (seg >= total) return;

    // locate (cluster, chunk-within-cluster) for this wave's segment
    int c = 0;
    while (chunkStart[c + 1] <= seg) ++c;
    const int j    = seg - chunkStart[c];
    const int base = rowStart[c] + (j << 4);
    const int cnt  = min(16, rowStart[c + 1] - base);

    const int m = lane & 15;   // A row / D column-group index
    const int h = lane >> 4;   // K-half selector

    // Fill weight K-pair table: W4[w][s*4+mm] = (W(2s,mm), W(2s+1,mm)); mm==3 or K>=319 -> 0
    const int wr0 = 3 * c;
    for (int e = lane; e < 640; e += 32) {
        int s  = e >> 2;
        int mm = e & 3;
        float x0 = 0.0f, x1 = 0.0f;
        if (mm < 3) {
            int row = wr0 + mm;
            int k0 = 2 * s, k1 = k0 + 1;
            x0 = (k0 < POS_SIZE) ? Wp[row * POS_SIZE + k0]
                                 : Wf[row * LATENT_SIZE + (k0 - POS_SIZE)];
            if (k1 < POS_SIZE)  x1 = Wp[row * POS_SIZE + k1];
            else if (k1 < KDIM) x1 = Wf[row * LATENT_SIZE + (k1 - POS_SIZE)];
        }
        v2f v; v.x = x0; v.y = x1;
        W4[w][e] = v;
    }
    // same-wave LDS RAW fence
    asm volatile("s_wait_dscnt 0" ::: "memory");

    // gather this lane's X row (duplicate last row for partial segments; stores are predicated)
    const int slot = min(m, cnt - 1);
    const int r = idx[base + slot];
    const float* __restrict__ Xr = X + (long)r * KDIM;

    // branch-free per-lane operand pointers
    const int mc = min(m, 3);                       // lanes m>=3 read the zero column (broadcast)
    const v2f* __restrict__ Ab = &W4[w][(h << 2) + mc];   // step t -> Ab[t*8]
    const float* __restrict__ Bp = Xr + 2 * h;            // step t -> Bp[4t], Bp[4t+1]

    v8f acc = {0.f, 0.f, 0.f, 0.f, 0.f, 0.f, 0.f, 0.f};

    // software-pipelined main loop: steps 0..77 (all indices <= 315, no guards)
    v2f a = Ab[0];
    v2f b; b.x = Bp[0]; b.y = Bp[1];
#pragma unroll 4
    for (int t = 0; t < 78; ++t) {
        v2f an = Ab[(t + 1) << 3];
        v2f bn; bn.x = Bp[4 * (t + 1)]; bn.y = Bp[4 * (t + 1) + 1];
        acc = __builtin_amdgcn_wmma_f32_16x16x4_f32(false, a, false, b,
                                                    (short)0, acc, false, false);
        a = an; b = bn;
    }
    // step 78 (operands already loaded)
    acc = __builtin_amdgcn_wmma_f32_16x16x4_f32(false, a, false, b,
                                                (short)0, acc, false, false);
    // step 79: K = 316..319; A col 319 is zero, clamp B index to stay in-bounds
    {
        v2f a79 = Ab[79 << 3];
        int kk = 316 + 2 * h;                       // <= 318
        v2f b79; b79.x = Xr[kk]; b79.y = Xr[min(kk + 1, KDIM - 1)];
        acc = __builtin_amdgcn_wmma_f32_16x16x4_f32(false, a79, false, b79,
                                                    (short)0, acc, false, false);
    }

    // lanes 0..15: acc[0..2] = rgb of x-row 'lane'
    if (h == 0 && m < cnt) {
        float* o = out + (long)r * 3;
        o[0] = acc[0];
        o[1] = acc[1];
        o[2] = acc[2];
    }
}

extern "C" void kernel_launch(void* const* d_in, const int* in_sizes, int n_in,
                              void* d_out, int out_size, void* d_ws, size_t ws_size,
                              hipStream_t stream) {
    const float* X   = (const float*)d_in[0];
    const int*   cid = (const int*)d_in[1];
    const float* Wp  = (const float*)d_in[2];
    const float* Wf  = (const float*)d_in[3];
    float* out = (float*)d_out;
    const int n = in_sizes[1];

    int* ws         = (int*)d_ws;
    int* cnt        = ws;          // 64
    int* rowStart   = ws + 64;     // 65
    int* chunkStart = ws + 129;    // 65
    int* cursor     = ws + 194;    // 64
    int* idx        = ws + 258;    // n

    const int eb = (n + 255) / 256;
    init_counts<<<1, 64, 0, stream>>>(cnt);
    hist_kernel<<<eb, 256, 0, stream>>>(cid, cnt, n);
    scan_kernel<<<1, 64, 0, stream>>>(cnt, rowStart, chunkStart, cursor);
    scatter_kernel<<<eb, 256, 0, stream>>>(cid, cursor, idx, n);

    const int segsUpper = (n + 15) / 16 + NUM_CLUSTERS;  // sum ceil(cnt_c/16) <= this
    const int cb = (segsUpper + CW - 1) / CW;
    compute_kernel<<<cb, 256, 0, stream>>>(X, Wp, Wf, rowStart, chunkStart, idx, out);
}